// GATLayer_84954453115117
// MI455X (gfx1250) — compile-verified
//
#include <hip/hip_runtime.h>

typedef __attribute__((ext_vector_type(2))) float v2f;
typedef __attribute__((ext_vector_type(8))) float v8f;

// ---------------------------------------------------------------------------
// Kernel 1: h = x @ W^T via V_WMMA_F32_16X16X4_F32.
//   x: [M,128] row-major, W: [128,128] row-major ([out,in]),  h: [M,128].
//   Block = 256 threads = 8 waves; each wave owns one 16-col tile (N=128),
//   block owns 16 rows. 32 K-steps of 4, accumulator chained in VGPRs.
//   A-frag layout (16x4 f32): lanes 0-15 -> K=k0+{0,1}; lanes 16-31 -> K=k0+{2,3}.
//   B-frag (4x16): same K split, N = lane%16.   C/D: VGPR r -> M=r+8*(lane/16).
// ---------------------------------------------------------------------------
__global__ __launch_bounds__(256) void gat_gemm_wmma(const float* __restrict__ x,
                                                     const float* __restrict__ W,
                                                     float* __restrict__ h) {
  const int lane = threadIdx.x & 31;
  const int wave = threadIdx.x >> 5;         // 0..7  -> N tile
  const int m0   = blockIdx.x << 4;          // 16 rows per block
  const int n0   = wave << 4;
  const int half = lane >> 4;                // 0: K lo pair, 1: K hi pair
  const int lrow = lane & 15;

  const float* __restrict__ xrow = x + (size_t)(m0 + lrow) * 128;
  const float* __restrict__ wrow = W + (size_t)(n0 + lrow) * 128;

  v8f acc = {};
#pragma unroll
  for (int k0 = 0; k0 < 128; k0 += 4) {
    const int kk = k0 + (half << 1);               // even -> 8B aligned
    v2f a = *(const v2f*)(xrow + kk);              // A: M=lrow, K=kk,kk+1
    v2f b = *(const v2f*)(wrow + kk);              // B: N=lrow, K=kk,kk+1
    acc = __builtin_amdgcn_wmma_f32_16x16x4_f32(
        /*neg_a=*/false, a, /*neg_b=*/false, b,
        /*c_mod=*/(short)0, acc, /*reuse_a=*/false, /*reuse_b=*/false);
  }

#pragma unroll
  for (int r = 0; r < 8; ++r) {
    h[(size_t)(m0 + r + (half << 3)) * 128 + n0 + lrow] = acc[r];
  }
}

// ---------------------------------------------------------------------------
// Kernel 2: per-node attention logits  al[n,h] = <h[n,h,:], a_left[h]>,
//                                      ar[n,h] = <h[n,h,:], a_right[h]>.
// One thread per (node, head); float4 inner loop over D=32.
// ---------------------------------------------------------------------------
__global__ void gat_node_scores(const float* __restrict__ h,
                                const float* __restrict__ a_left,
                                const float* __restrict__ a_right,
                                float* __restrict__ al, float* __restrict__ ar,
                                int total) {
  int t = blockIdx.x * blockDim.x + threadIdx.x;
  if (t >= total) return;
  const int n  = t >> 2;
  const int hh = t & 3;
  const float4* __restrict__ hp = (const float4*)(h + (size_t)n * 128 + hh * 32);
  const float4* __restrict__ wl = (const float4*)(a_left  + hh * 32);
  const float4* __restrict__ wr = (const float4*)(a_right + hh * 32);
  float sl = 0.f, sr = 0.f;
#pragma unroll
  for (int i = 0; i < 8; ++i) {
    float4 v = hp[i], l = wl[i], r = wr[i];
    sl += v.x * l.x + v.y * l.y + v.z * l.z + v.w * l.w;
    sr += v.x * r.x + v.y * r.y + v.z * r.z + v.w * r.w;
  }
  al[t] = sl;
  ar[t] = sr;
}

// ---------------------------------------------------------------------------
// Kernel 3: zero out[] (N*128) and denom[] (N*4) — harness poisons buffers.
// ---------------------------------------------------------------------------
__global__ void gat_zero(float* __restrict__ out, float* __restrict__ denom,
                         int n_out, int n_denom) {
  int t = blockIdx.x * blockDim.x + threadIdx.x;
  if (t < n_out) out[t] = 0.f;
  if (t < n_denom) denom[t] = 0.f;
}

// ---------------------------------------------------------------------------
// Kernel 4: per-edge  e = leaky_relu(al[src]+ar[dst]); e_exp = exp(e);
//           denom[dst,h] += e_exp  (atomic, L2-resident).
// One thread per edge; 128-bit loads of the 4-head score vectors.
// ---------------------------------------------------------------------------
__global__ void gat_edge_softmax_num(const long long* __restrict__ src,
                                     const long long* __restrict__ dst,
                                     const float4* __restrict__ al,
                                     const float4* __restrict__ ar,
                                     float4* __restrict__ e_exp,
                                     float* __restrict__ denom, int E) {
  int e = blockIdx.x * blockDim.x + threadIdx.x;
  if (e >= E) return;
  const int s = (int)src[e];
  const int d = (int)dst[e];
  float4 va = al[s];
  float4 vb = ar[d];
  float4 ex;
  float v;
  v = va.x + vb.x; v = v > 0.f ? v : 0.2f * v; ex.x = expf(v);
  v = va.y + vb.y; v = v > 0.f ? v : 0.2f * v; ex.y = expf(v);
  v = va.z + vb.z; v = v > 0.f ? v : 0.2f * v; ex.z = expf(v);
  v = va.w + vb.w; v = v > 0.f ? v : 0.2f * v; ex.w = expf(v);
  e_exp[e] = ex;
  float* dn = denom + (size_t)d * 4;
  atomicAdd(dn + 0, ex.x);
  atomicAdd(dn + 1, ex.y);
  atomicAdd(dn + 2, ex.z);
  atomicAdd(dn + 3, ex.w);
}

// ---------------------------------------------------------------------------
// Kernel 5: out[dst, h, d] += (e_exp[e,h] / (denom[dst,h]+1e-8)) * hmat[src,h,d]
// One thread per (edge, j) with j in [0,128); consecutive threads hit
// contiguous addresses of h[src] / out[dst] -> coalesced gather + atomics.
// ---------------------------------------------------------------------------
__global__ void gat_edge_aggregate(const long long* __restrict__ src,
                                   const long long* __restrict__ dst,
                                   const float* __restrict__ hmat,
                                   const float* __restrict__ e_exp,
                                   const float* __restrict__ denom,
                                   float* __restrict__ out, long long total) {
  long long t = (long long)blockIdx.x * blockDim.x + threadIdx.x;
  if (t >= total) return;
  const int e  = (int)(t >> 7);
  const int j  = (int)(t & 127);
  const int hh = j >> 5;
  const int s = (int)src[e];
  const int d = (int)dst[e];
  const float alpha = e_exp[(size_t)e * 4 + hh] / (denom[(size_t)d * 4 + hh] + 1e-8f);
  atomicAdd(out + (size_t)d * 128 + j, alpha * hmat[(size_t)s * 128 + j]);
}

// ---------------------------------------------------------------------------
extern "C" void kernel_launch(void* const* d_in, const int* in_sizes, int n_in,
                              void* d_out, int out_size, void* d_ws, size_t ws_size,
                              hipStream_t stream) {
  const float*     x      = (const float*)d_in[0];
  const long long* eidx   = (const long long*)d_in[1];   // int64 [2, E]
  const float*     W      = (const float*)d_in[2];
  const float*     a_l    = (const float*)d_in[3];
  const float*     a_r    = (const float*)d_in[4];
  float*           out    = (float*)d_out;

  const int N = in_sizes[0] / 128;   // 100000
  const int E = in_sizes[1] / 2;     // 1600000

  // Workspace layout (bytes), all 16B-aligned:
  char* ws = (char*)d_ws;
  float*  h     = (float*)(ws);                                    // N*128 f32
  float*  al    = (float*)(ws + (size_t)N * 128 * 4);              // N*4
  float*  ar    = (float*)(ws + (size_t)N * 132 * 4);              // N*4
  float*  denom = (float*)(ws + (size_t)N * 136 * 4);              // N*4
  float*  e_exp = (float*)(ws + (size_t)N * 140 * 4);              // E*4

  const long long* src = eidx;        // row 0
  const long long* dst = eidx + E;    // row 1

  // 1) WMMA GEMM: h = x @ W^T   (M=N_nodes multiple of 16 -> exact tiling)
  gat_gemm_wmma<<<N / 16, 256, 0, stream>>>(x, W, h);

  // 2) Per-node attention logits
  gat_node_scores<<<(N * 4 + 255) / 256, 256, 0, stream>>>(h, a_l, a_r, al, ar, N * 4);

  // 3) Zero out + denom
  gat_zero<<<(N * 128 + 255) / 256, 256, 0, stream>>>(out, denom, N * 128, N * 4);

  // 4) Edge softmax numerators + denominator accumulation
  gat_edge_softmax_num<<<(E + 255) / 256, 256, 0, stream>>>(
      src, dst, (const float4*)al, (const float4*)ar, (float4*)e_exp, denom, E);

  // 5) Weighted scatter-aggregate
  const long long total = (long long)E * 128;
  gat_edge_aggregate<<<(unsigned)((total + 255) / 256), 256, 0, stream>>>(
      src, dst, h, e_exp, denom, out, total);
}